// TtLlamaAttention_15212774162885
// MI455X (gfx1250) — compile-verified
//
#include <hip/hip_runtime.h>
#include <math.h>

// Llama decode attention step, MI455X (gfx1250, wave32).
// Memory-bound (~270 MB streamed vs ~0.5 GFLOP, floor ~11.6us @ 23.3 TB/s):
// everything is structured as coalesced streaming. The two fp32 GEMMs use
// V_WMMA_F32_16X16X4_F32 with split-K for occupancy; attention is a
// split-softmax flash-decode with float4 row streaming.

#define B_      4
#define H_      32
#define KVH_    8
#define D_      128
#define HID_    4096
#define SW_     4096
#define G_      4               // H_/KVH_
#define NQKV_   6144            // (H_+2*KVH_)*D_
#define NCHUNK_ 32
#define CHUNK_  128             // SW_/NCHUNK_
#define KSPLIT_ 8
#define KCHUNK_ 512             // HID_/KSPLIT_

typedef float v2f __attribute__((ext_vector_type(2)));
typedef float v8f __attribute__((ext_vector_type(8)));

// ---------------------------------------------------------------------------
// Split-K GEMM: part(kc, 4 x N) = x(4 x [kc-slice]) @ w([kc-slice] x N).
// One wave owns one 16-column tile; 4 waves per block; blockIdx.y = K-chunk.
// x slice is staged in LDS once (branch-free masked reads afterwards).
// A-frag (16x4 f32): lane 0-15 -> {K=0,K=1}, lane 16-31 -> {K=2,K=3},
//                    row M = lane & 15 (rows 4..15 zeroed via mask mult).
// B-frag (4x16 f32): col N = lane & 15; the two VGPRs hold rows {kk, kk+1},
//                    kk = (lane>=16) ? 2 : 0 (mirrors A layout).
// C/D (16x16 f32):   VGPR m, lanes 0-15 -> row M=m, col N=lane.
// ---------------------------------------------------------------------------
template <int N>
__global__ __launch_bounds__(128) void k_gemm(const float* __restrict__ x,
                                              const float* __restrict__ w,
                                              float* __restrict__ part) {
  __shared__ float sx[B_ * KCHUNK_];   // 8 KB
  const int tid   = threadIdx.x;
  const int lane  = tid & 31;
  const int wv    = tid >> 5;
  const int j0    = (blockIdx.x * 4 + wv) * 16;
  const int kbase = blockIdx.y * KCHUNK_;
  const int row   = lane & 15;
  const int col   = lane & 15;
  const int kk    = (lane >> 4) << 1;          // 0 or 2
  const float msk = (row < B_) ? 1.f : 0.f;    // zero-pad rows 4..15
  const int srow  = (row & 3) * KCHUNK_;

  // Cooperative stage of the 4 x 512 x-slice into LDS (float4 stores).
  for (int t = tid; t < (B_ * KCHUNK_) / 4; t += 128) {
    const int r = t >> 7;               // /(KCHUNK_/4)
    const int o = (t & 127) * 4;
    const float4 v = *reinterpret_cast<const float4*>(x + (size_t)r * HID_ + kbase + o);
    *reinterpret_cast<float4*>(&sx[r * KCHUNK_ + o]) = v;
  }
  __syncthreads();

  v8f acc = {0.f, 0.f, 0.f, 0.f, 0.f, 0.f, 0.f, 0.f};

  for (int k = 0; k < KCHUNK_; k += 4) {
    v2f a, b;
    a.x = msk * sx[srow + k + kk];
    a.y = msk * sx[srow + k + kk + 1];
    const size_t wrow = (size_t)(kbase + k + kk) * N + j0 + col;
    b.x = w[wrow];
    b.y = w[wrow + N];
    __builtin_prefetch(w + wrow + (size_t)32 * N, 0, 1);  // global_prefetch_b8
    acc = __builtin_amdgcn_wmma_f32_16x16x4_f32(
        /*neg_a=*/false, a, /*neg_b=*/false, b,
        /*c_mod=*/(short)0, acc, /*reuse_a=*/false, /*reuse_b=*/false);
  }

  if (lane < 16) {
#pragma unroll
    for (int m = 0; m < B_; ++m)
      part[((size_t)blockIdx.y * B_ + m) * N + j0 + lane] = acc[m];
  }
}

// Deterministic fixed-order reduction over the KSPLIT_ partials.
__global__ __launch_bounds__(256) void k_reduce(const float* __restrict__ part,
                                                float* __restrict__ out, int n) {
  const int i = blockIdx.x * 256 + threadIdx.x;
  if (i < n) {
    float s = 0.f;
#pragma unroll
    for (int c = 0; c < KSPLIT_; ++c) s += part[(size_t)c * n + i];
    out[i] = s;
  }
}

// ---------------------------------------------------------------------------
// Rotary: per head, row(128) @ rot(128x128). One block per (b, head).
// ---------------------------------------------------------------------------
__global__ __launch_bounds__(128) void k_rot(const float* __restrict__ qkv,
                                             const float* __restrict__ rot,
                                             float* __restrict__ qrot,
                                             float* __restrict__ krot) {
  __shared__ float sx[D_];
  const int bid = blockIdx.x;
  const int b   = bid / (H_ + KVH_);
  const int hh  = bid % (H_ + KVH_);
  const float* src;
  float* dst;
  if (hh < H_) {
    src = qkv + (size_t)b * NQKV_ + hh * D_;
    dst = qrot + ((size_t)b * H_ + hh) * D_;
  } else {
    const int kvh = hh - H_;
    src = qkv + (size_t)b * NQKV_ + H_ * D_ + kvh * D_;
    dst = krot + ((size_t)b * KVH_ + kvh) * D_;
  }
  const int e = threadIdx.x;
  sx[e] = src[e];
  __syncthreads();
  float acc = 0.f;
#pragma unroll 8
  for (int d = 0; d < D_; ++d) acc += sx[d] * rot[d * D_ + e];
  dst[e] = acc;
}

// ---------------------------------------------------------------------------
// Flash-decode attention partials. Grid (chunk, kvh, b); 8 waves per block,
// each wave walks CHUNK_/8 positions; whole wave cooperates per position:
// coalesced float4 K/V row loads (512B per row per wave) + butterfly reduce.
// Position s == current_pos substitutes the freshly rotated k / new v (the
// cache inputs are never mutated, matching the functional reference).
// ---------------------------------------------------------------------------
__global__ __launch_bounds__(256) void k_attn(const float* __restrict__ qrot,
                                              const float* __restrict__ krot,
                                              const float* __restrict__ qkv,
                                              const float* __restrict__ cache_k,
                                              const float* __restrict__ cache_v,
                                              const int* __restrict__ cur_pos,
                                              float* __restrict__ pm,
                                              float* __restrict__ pl,
                                              float* __restrict__ pacc) {
  const int chunk = blockIdx.x;
  const int kvh   = blockIdx.y;
  const int b     = blockIdx.z;
  const int lane  = threadIdx.x & 31;
  const int wv    = threadIdx.x >> 5;
  const int cur   = cur_pos[0];
  const float scale = 0.08838834764831845f;  // 1/sqrt(128)

  float4 qv[G_];
#pragma unroll
  for (int g = 0; g < G_; ++g)
    qv[g] = reinterpret_cast<const float4*>(
        qrot + ((size_t)b * H_ + kvh * G_ + g) * D_)[lane];

  float  m[G_], l[G_];
  float4 a[G_];
#pragma unroll
  for (int g = 0; g < G_; ++g) {
    m[g] = -1e30f;
    l[g] = 0.f;
    a[g] = make_float4(0.f, 0.f, 0.f, 0.f);
  }

  const int npos = CHUNK_ / 8;          // positions per wave
  const int s0   = chunk * CHUNK_ + wv * npos;
  for (int i = 0; i < npos; ++i) {
    const int s = s0 + i;
    if (s > cur) break;  // uniform across the wave
    const float* krow;
    const float* vrow;
    if (s == cur) {
      krow = krot + ((size_t)b * KVH_ + kvh) * D_;
      vrow = qkv + (size_t)b * NQKV_ + (H_ + KVH_) * D_ + kvh * D_;
    } else {
      const size_t base = (((size_t)b * KVH_ + kvh) * SW_ + s) * D_;
      krow = cache_k + base;
      vrow = cache_v + base;
    }
    const float4 kk = reinterpret_cast<const float4*>(krow)[lane];
    const float4 vv = reinterpret_cast<const float4*>(vrow)[lane];
#pragma unroll
    for (int g = 0; g < G_; ++g) {
      float p = qv[g].x * kk.x + qv[g].y * kk.y + qv[g].z * kk.z + qv[g].w * kk.w;
#pragma unroll
      for (int off = 16; off > 0; off >>= 1) p += __shfl_xor(p, off, 32);
      const float sc   = p * scale;
      const float mn   = fmaxf(m[g], sc);
      const float corr = __expf(m[g] - mn);
      const float wgt  = __expf(sc - mn);
      l[g] = l[g] * corr + wgt;
      a[g].x = a[g].x * corr + wgt * vv.x;
      a[g].y = a[g].y * corr + wgt * vv.y;
      a[g].z = a[g].z * corr + wgt * vv.z;
      a[g].w = a[g].w * corr + wgt * vv.w;
      m[g] = mn;
    }
  }

  // Intra-block (8-wave) combine via LDS.
  __shared__ float sm[8][G_];
  __shared__ float sl[8][G_];
  __shared__ float sa[8][G_][D_];
  if (lane == 0) {
#pragma unroll
    for (int g = 0; g < G_; ++g) { sm[wv][g] = m[g]; sl[wv][g] = l[g]; }
  }
#pragma unroll
  for (int g = 0; g < G_; ++g) {
    sa[wv][g][lane * 4 + 0] = a[g].x;
    sa[wv][g][lane * 4 + 1] = a[g].y;
    sa[wv][g][lane * 4 + 2] = a[g].z;
    sa[wv][g][lane * 4 + 3] = a[g].w;
  }
  __syncthreads();

  if (threadIdx.x < D_) {
    const int d = threadIdx.x;
#pragma unroll
    for (int g = 0; g < G_; ++g) {
      float M = -1e30f;
      for (int w = 0; w < 8; ++w) M = fmaxf(M, sm[w][g]);
      float L = 0.f, A = 0.f;
      for (int w = 0; w < 8; ++w) {
        const float e = __expf(sm[w][g] - M);
        L += sl[w][g] * e;
        A += sa[w][g][d] * e;
      }
      const size_t idx = ((size_t)(b * KVH_ + kvh) * G_ + g) * NCHUNK_ + chunk;
      pacc[idx * D_ + d] = A;
      if (d == 0) { pm[idx] = M; pl[idx] = L; }
    }
  }
}

// ---------------------------------------------------------------------------
// Combine chunk partials -> ctx (B, H*D), laid out as (b,kvh,g,d) which is
// exactly the reference's reshape ordering.
// ---------------------------------------------------------------------------
__global__ __launch_bounds__(128) void k_comb(const float* __restrict__ pm,
                                              const float* __restrict__ pl,
                                              const float* __restrict__ pacc,
                                              float* __restrict__ ctx) {
  const int bg = blockIdx.x;    // (b*KVH + kvh)*G + g
  const int d  = threadIdx.x;
  float M = -1e30f;
  for (int c = 0; c < NCHUNK_; ++c) M = fmaxf(M, pm[(size_t)bg * NCHUNK_ + c]);
  float L = 0.f, A = 0.f;
  for (int c = 0; c < NCHUNK_; ++c) {
    const float e = __expf(pm[(size_t)bg * NCHUNK_ + c] - M);
    L += pl[(size_t)bg * NCHUNK_ + c] * e;
    A += pacc[((size_t)bg * NCHUNK_ + c) * D_ + d] * e;
  }
  ctx[(size_t)bg * D_ + d] = A / L;
}

// ---------------------------------------------------------------------------
extern "C" void kernel_launch(void* const* d_in, const int* in_sizes, int n_in,
                              void* d_out, int out_size, void* d_ws, size_t ws_size,
                              hipStream_t stream) {
  const float* x    = (const float*)d_in[0];
  const float* wqkv = (const float*)d_in[1];
  const float* wo   = (const float*)d_in[2];
  const float* rot  = (const float*)d_in[3];
  const float* ck   = (const float*)d_in[4];
  const float* cv   = (const float*)d_in[5];
  const int*   cur  = (const int*)d_in[6];

  float* ws   = (float*)d_ws;
  float* qkv  = ws;                                    // 24576
  float* qrot = qkv  + (size_t)B_ * NQKV_;             // 16384
  float* krot = qrot + (size_t)B_ * H_ * D_;           // 4096
  float* pm   = krot + (size_t)B_ * KVH_ * D_;         // 4096
  float* pl   = pm   + (size_t)B_ * KVH_ * G_ * NCHUNK_;
  float* pacc = pl   + (size_t)B_ * KVH_ * G_ * NCHUNK_;      // 524288
  float* ctx  = pacc + (size_t)B_ * KVH_ * G_ * NCHUNK_ * D_; // 16384
  float* pqkv = ctx  + (size_t)B_ * H_ * D_;           // 8*24576
  float* pout = pqkv + (size_t)KSPLIT_ * B_ * NQKV_;   // 8*16384

  // 1) QKV projection: split-K WMMA GEMM -> partials -> reduce.
  k_gemm<NQKV_><<<dim3(NQKV_ / 64, KSPLIT_), 128, 0, stream>>>(x, wqkv, pqkv);
  k_reduce<<<(B_ * NQKV_ + 255) / 256, 256, 0, stream>>>(pqkv, qkv, B_ * NQKV_);

  // 2) Rotary for q and k heads.
  k_rot<<<B_ * (H_ + KVH_), 128, 0, stream>>>(qkv, rot, qrot, krot);

  // 3) Flash-decode partials (1024 blocks to widen the KV-cache stream).
  dim3 g3(NCHUNK_, KVH_, B_);
  k_attn<<<g3, 256, 0, stream>>>(qrot, krot, qkv, ck, cv, cur, pm, pl, pacc);

  // 4) Combine partials -> ctx.
  k_comb<<<B_ * H_, 128, 0, stream>>>(pm, pl, pacc, ctx);

  // 5) Output projection: split-K WMMA GEMM -> partials -> reduce into d_out.
  k_gemm<HID_><<<dim3(HID_ / 64, KSPLIT_), 128, 0, stream>>>(ctx, wo, pout);
  k_reduce<<<(B_ * HID_ + 255) / 256, 256, 0, stream>>>(pout, (float*)d_out, B_ * HID_);
}